// QuantumGate_65481071410301
// MI455X (gfx1250) — compile-verified
//
#include <hip/hip_runtime.h>
#include <math.h>

typedef float v2f __attribute__((ext_vector_type(2)));
typedef float v8f __attribute__((ext_vector_type(8)));

#define NQ   14
#define DIM  16384      // 2^14 amplitudes
#define NT   256        // 8 wave32s per workgroup

// Wrap the CDNA5 WMMA builtin so the host pass never parses it.
// __gfx1250__ is defined only during the gfx1250 device compilation.
static __device__ inline v8f wmma_16x16x4_f32(v2f a, v2f b, v8f c) {
#if defined(__gfx1250__)
  // 8 args: (neg_a, A, neg_b, B, c_mod, C, reuse_a, reuse_b)
  return __builtin_amdgcn_wmma_f32_16x16x4_f32(
      false, a, false, b, (short)0, c, false, false);
#else
  return c;  // host-pass stub; never executed on device
#endif
}

// One workgroup simulates one batch sample entirely in LDS.
// State layout: st[0..DIM) = real, st[DIM..2*DIM) = imag (fp32, SoA).
__global__ __launch_bounds__(NT) void qsim14_kernel(
    const float* __restrict__ x,       // (B, 14) RX angles
    const float* __restrict__ params,  // (28,)  RY angles, p[layer*14 + w]
    float* __restrict__ out)           // (B, 14) Z expectations
{
  __shared__ float st[2 * DIM];        // 128 KB state
  __shared__ float pHi[128], pLo[128]; // product-state half factors
  __shared__ float cq[NQ], sq[NQ];     // cos/sin(x/2) per qubit
  __shared__ float red[NQ * 8];        // cross-wave reduction scratch

  const int b    = blockIdx.x;
  const int tid  = threadIdx.x;
  const int lane = tid & 31;
  const int wave = tid >> 5;

  // ------------------------------------------------------------------
  // Phase 0: RX(x) product state.  amp(idx) = (-i)^popcount(idx) * prod(mag)
  // qubit q lives at state bit (13 - q).
  // ------------------------------------------------------------------
  if (tid < NQ) {
    float h = 0.5f * x[b * NQ + tid];
    cq[tid] = cosf(h);
    sq[tid] = sinf(h);
  }
  __syncthreads();

  if (tid < 128) {           // high 7 bits: qubits 0..6 <-> h bits 6..0
    int h = tid; float p = 1.f;
#pragma unroll
    for (int t = 0; t < 7; ++t) p *= ((h >> (6 - t)) & 1) ? sq[t] : cq[t];
    pHi[h] = p;
  } else {                   // low 7 bits: qubits 7..13 <-> l bits 6..0
    int l = tid - 128; float p = 1.f;
#pragma unroll
    for (int u = 0; u < 7; ++u) p *= ((l >> (6 - u)) & 1) ? sq[7 + u] : cq[7 + u];
    pLo[l] = p;
  }
  __syncthreads();

#pragma unroll 4
  for (int k = 0; k < 64; ++k) {
    int idx = k * NT + tid;
    float m = pHi[idx >> 7] * pLo[idx & 127];
    int ph = __popc(idx) & 3;                 // (-i)^ph
    st[idx]       = (ph == 0) ? m : ((ph == 2) ? -m : 0.f);
    st[DIM + idx] = (ph == 1) ? -m : ((ph == 3) ? m : 0.f);
  }
  __syncthreads();

  // ------------------------------------------------------------------
  // Phase 1: two layers of [grouped RY via WMMA]  +  [CNOT chain = Gray gather]
  // ------------------------------------------------------------------
  const int baseArr[4] = {10, 6, 2, 0};

  for (int L = 0; L < 2; ++L) {
    for (int g = 0; g < 4; ++g) {
      const int base = baseArr[g];
      const int mask = (1 << base) - 1;

      // 16x16 group gate G = tensor product of 4 (R or I) factors.
      // mid bit m (0=LSB) corresponds to qubit (13 - base - m);
      // group 3 applies identity on its top two slots (qubits 10,11).
      float gc[4], gs[4];
#pragma unroll
      for (int m = 0; m < 4; ++m) {
        if (g == 3 && m >= 2) { gc[m] = 1.f; gs[m] = 0.f; }
        else {
          int q = 13 - base - m;
          float h = 0.5f * params[L * NQ + q];
          gc[m] = cosf(h); gs[m] = sinf(h);
        }
      }

      // A-fragments (16x4 f32 layout): VGPR r, lane l -> G[l&15][4k + 2*(l>>4) + r]
      const int M    = lane & 15;
      const int Koff = (lane >> 4) << 1;
      v2f A[4];
#pragma unroll
      for (int k = 0; k < 4; ++k) {
#pragma unroll
        for (int r = 0; r < 2; ++r) {
          int Kc = 4 * k + Koff + r;
          float e = 1.f;
#pragma unroll
          for (int m = 0; m < 4; ++m) {
            int ib = (M >> m) & 1, jb = (Kc >> m) & 1;
            e *= (ib == jb) ? gc[m] : ((ib == 0) ? -gs[m] : gs[m]);
          }
          A[k][r] = e;
        }
      }

      // 64 column tiles of 16; each wave owns disjoint tiles -> in-place safe.
      for (int t = wave; t < 64; t += 8) {
        int c = 16 * t + (lane & 15);
        int colAddr = ((c >> base) << (base + 4)) + (c & mask);
        int rowsel  = (lane >> 4) << 1;   // B 4x16 layout: K = r + 2*(l>>4)
        int Moff    = (lane >> 4) << 3;   // D 16x16 layout: M = r + 8*(l>>4)
#pragma unroll
        for (int comp = 0; comp < 2; ++comp) {
          float* buf = st + comp * DIM;
          v8f acc = {};
#pragma unroll
          for (int k = 0; k < 4; ++k) {
            v2f Bf;
            Bf[0] = buf[colAddr + ((4 * k + rowsel + 0) << base)];
            Bf[1] = buf[colAddr + ((4 * k + rowsel + 1) << base)];
            acc = wmma_16x16x4_f32(A[k], Bf, acc);
          }
#pragma unroll
          for (int r = 0; r < 8; ++r)
            buf[colAddr + ((r + Moff) << base)] = acc[r];
        }
      }
      __syncthreads();
    }

    // CNOT chain w=0..12 collapses to: new[j] = old[j ^ (j>>1)] (Gray gather).
    // Register-stage a full component per thread (64 floats) to permute in place.
#pragma unroll
    for (int comp = 0; comp < 2; ++comp) {
      float* buf = st + comp * DIM;
      float v[64];
#pragma unroll
      for (int k = 0; k < 64; ++k) {
        int j = k * NT + tid;
        v[k] = buf[j ^ (j >> 1)];
      }
      __syncthreads();
#pragma unroll
      for (int k = 0; k < 64; ++k)
        buf[k * NT + tid] = v[k];
      __syncthreads();
    }
  }

  // ------------------------------------------------------------------
  // Phase 2: Z expectations. out[b][w] = sum_idx (1 - 2*bit(idx,13-w)) * |amp|^2
  // ------------------------------------------------------------------
  float acc14[NQ];
#pragma unroll
  for (int w = 0; w < NQ; ++w) acc14[w] = 0.f;

  for (int k = 0; k < 64; ++k) {
    int j = k * NT + tid;
    float re = st[j], im = st[DIM + j];
    float p = re * re + im * im;
#pragma unroll
    for (int w = 0; w < NQ; ++w)
      acc14[w] += ((j >> (13 - w)) & 1) ? -p : p;
  }

#pragma unroll
  for (int w = 0; w < NQ; ++w) {
    float v = acc14[w];
#pragma unroll
    for (int off = 16; off > 0; off >>= 1)
      v += __shfl_down(v, off, 32);
    if (lane == 0) red[w * 8 + wave] = v;
  }
  __syncthreads();
  if (tid < NQ) {
    float s = 0.f;
#pragma unroll
    for (int i = 0; i < 8; ++i) s += red[tid * 8 + i];
    out[b * NQ + tid] = s;
  }
}

extern "C" void kernel_launch(void* const* d_in, const int* in_sizes, int n_in,
                              void* d_out, int out_size, void* d_ws, size_t ws_size,
                              hipStream_t stream) {
  const float* x      = (const float*)d_in[0];   // (B, 14) fp32
  const float* params = (const float*)d_in[1];   // (28,)   fp32
  float* out          = (float*)d_out;           // (B, 14) fp32
  int B = in_sizes[0] / NQ;
  qsim14_kernel<<<dim3(B), dim3(NT), 0, stream>>>(x, params, out);
}